// node_RGCN_39668317946546
// MI455X (gfx1250) — compile-verified
//
#include <hip/hip_runtime.h>
#include <hip/hip_bf16.h>

// ---------------- problem constants (match reference) ----------------
#define RG_N      40000      // nodes
#define RG_E      640000     // edges
#define RG_R      8          // relations
#define RG_B      4          // bases
#define RG_D      256        // embedding dim
#define RG_RN     (RG_R * RG_N)

typedef __attribute__((ext_vector_type(2))) float v2f;
typedef __attribute__((ext_vector_type(8))) float v8f;

// ---------------- small utility kernels ----------------

__global__ void rgcn_zero_f4(float4* __restrict__ p, long n4) {
    long i = (long)blockIdx.x * blockDim.x + threadIdx.x;
    long stride = (long)gridDim.x * blockDim.x;
    for (; i < n4; i += stride) p[i] = make_float4(0.f, 0.f, 0.f, 0.f);
}

__global__ void rgcn_count_edges(const int* __restrict__ ei,
                                 const int* __restrict__ et,
                                 float* __restrict__ cnt) {
    int e = blockIdx.x * blockDim.x + threadIdx.x;
    if (e < RG_E) {
        int dst = ei[RG_E + e];
        long seg = (long)et[e] * RG_N + dst;
        atomicAdd(&cnt[seg], 1.0f);
    }
}

__global__ void rgcn_invert_cnt(float* __restrict__ cnt) {
    int i = blockIdx.x * blockDim.x + threadIdx.x;
    if (i < RG_RN) cnt[i] = 1.0f / fmaxf(cnt[i], 1.0f);
}

// W[r,i,o] = sum_b comp[r,b] * basis[b,i,o]   (one layer's slice)
__global__ void rgcn_compute_w(const float* __restrict__ comp,   // [R,B]
                               const float* __restrict__ basis,  // [B,D,D]
                               float* __restrict__ W) {          // [R,D,D]
    int idx = blockIdx.x * blockDim.x + threadIdx.x;   // < R*D*D
    int r  = idx >> 16;          // D*D == 65536
    int io = idx & 65535;
    float s = 0.f;
#pragma unroll
    for (int b = 0; b < RG_B; ++b)
        s += comp[r * RG_B + b] * basis[(long)b * RG_D * RG_D + io];
    W[idx] = s;
}

// scatter-add: agg[(etype*N + dst)*D + d] += x[src*D + d]
__global__ void rgcn_scatter(const int* __restrict__ ei,
                             const int* __restrict__ et,
                             const float* __restrict__ x,
                             float* __restrict__ agg) {
    int e = blockIdx.x;
    int d = threadIdx.x;
    int src = ei[e];
    int dst = ei[RG_E + e];
    long seg = (long)et[e] * RG_N + dst;
    atomicAdd(&agg[seg * RG_D + d], x[(long)src * RG_D + d]);
}

__global__ void rgcn_relu(float* __restrict__ h, long n) {
    long i = (long)blockIdx.x * blockDim.x + threadIdx.x;
    long stride = (long)gridDim.x * blockDim.x;
    for (; i < n; i += stride) h[i] = fmaxf(h[i], 0.f);
}

// ---------------- fused RGCN GEMM (WMMA f32 16x16x4) ----------------
// out[n,:] = sum_{r<8} (agg[r,n,:] * invcnt[r*N+n]) @ W[r] + x[n,:] @ root + bias
//
// Block: 256 threads = 8 wave32. Block tile: 256 (M) x 64 (N).
// Each wave: 32 (M) x 64 (N) = 8 x v8f accumulators.
// K loop over 9 source matrices x 256, staged through LDS in chunks of 32.
//
// LDS layouts:
//   As: row-major, padded stride 36 floats (bank-conflict-free b64 frag loads,
//       16B-aligned b128 staging stores).
//   Bs: fragment-major -- fragment (k4/4, ns) occupies 64 consecutive floats,
//       lane L's (v0,v1) pair at frag*64 + L*2. Fragment load = one b64,
//       256B contiguous per wave (all 64 banks hit exactly once).
#define AT_STRIDE 36

__global__ void __launch_bounds__(256)
rgcn_gemm(const float* __restrict__ agg,    // [R*N, D] (unscaled sums)
          const float* __restrict__ x,      // [N, D]   (root-term input)
          const float* __restrict__ Wrel,   // [R, D, D]
          const float* __restrict__ root,   // [D, D]
          const float* __restrict__ bias,   // [D]
          const float* __restrict__ invcnt, // [R*N]
          float* __restrict__ out) {        // [N, D]
    __shared__ float As[256 * AT_STRIDE];   // 36 KB
    __shared__ float Bs[32 * 64];           //  8 KB (frag-major)

    const int lane = threadIdx.x & 31;
    const int wave = threadIdx.x >> 5;
    const int half = lane >> 4;         // 0: lanes 0-15, 1: lanes 16-31
    const int l15  = lane & 15;
    const int blockM = blockIdx.x * 256;
    const int blockN = blockIdx.y * 64;

    v8f acc0[4] = {};   // rows wave*32 + 0..15, 4 n-subtiles
    v8f acc1[4] = {};   // rows wave*32 + 16..31

    for (int m = 0; m < 9; ++m) {
        const float* Asrc = (m < 8) ? (agg + (long)m * RG_N * RG_D) : x;
        const float* Bsrc = (m < 8) ? (Wrel + (long)m * RG_D * RG_D) : root;
        const float* inv  = (m < 8) ? (invcnt + (long)m * RG_N) : nullptr;

        for (int kk = 0; kk < RG_D; kk += 32) {
            __syncthreads();   // previous chunk fully consumed
            // ---- stage A: 256 rows x 32 k (scaled by inv_cnt) ----
#pragma unroll
            for (int i = 0; i < 8; ++i) {
                int f4  = threadIdx.x + i * 256;   // < 2048 float4s
                int row = f4 >> 3;                 // 8 float4 per row
                int col = (f4 & 7) << 2;
                int grow = blockM + row;
                float4 v = make_float4(0.f, 0.f, 0.f, 0.f);
                if (grow < RG_N) {
                    v = *(const float4*)(Asrc + (long)grow * RG_D + kk + col);
                    if (inv) {
                        float s = inv[grow];
                        v.x *= s; v.y *= s; v.z *= s; v.w *= s;
                    }
                }
                *(float4*)(&As[row * AT_STRIDE + col]) = v;
            }
            // ---- stage B: 32 k x 64 n, scattered into fragment order ----
#pragma unroll
            for (int i = 0; i < 2; ++i) {
                int f4  = threadIdx.x + i * 256;   // < 512 float4s
                int row = f4 >> 4;                 // k row 0..31
                int col = (f4 & 15) << 2;          // n col 0..60
                float4 v = *(const float4*)(Bsrc + (long)(kk + row) * RG_D
                                            + blockN + col);
                int fbase = ((row >> 2) * 4 + (col >> 4)) * 64;
                int l0 = (col & 15) + ((row & 3) >> 1) * 16;  // lane of v.x
                int vv = row & 1;                              // v0 or v1
                Bs[fbase + (l0 + 0) * 2 + vv] = v.x;
                Bs[fbase + (l0 + 1) * 2 + vv] = v.y;
                Bs[fbase + (l0 + 2) * 2 + vv] = v.z;
                Bs[fbase + (l0 + 3) * 2 + vv] = v.w;
            }
            __syncthreads();
            // ---- compute: 8 k-quads, 4 n-subtiles x 2 m-subtiles each ----
#pragma unroll
            for (int k4 = 0; k4 < 32; k4 += 4) {
                const int kb = k4 + (half << 1);   // per-lane K base (ISA layout)
                v2f a0 = *(const v2f*)(&As[(wave * 32 + l15) * AT_STRIDE + kb]);
                v2f a1 = *(const v2f*)(&As[(wave * 32 + 16 + l15) * AT_STRIDE + kb]);
#pragma unroll
                for (int ns = 0; ns < 4; ++ns) {
                    v2f b = *(const v2f*)(&Bs[((k4 >> 2) * 4 + ns) * 64 + lane * 2]);
                    acc0[ns] = __builtin_amdgcn_wmma_f32_16x16x4_f32(
                        false, a0, false, b, (short)0, acc0[ns], false, false);
                    acc1[ns] = __builtin_amdgcn_wmma_f32_16x16x4_f32(
                        false, a1, false, b, (short)0, acc1[ns], false, false);
                }
            }
        }
    }

    // ---- store: C/D layout: VGPR j -> rows (j, j+8), lane halves split M, lane%16 -> N ----
    const int rowBase0 = blockM + wave * 32 + (half << 3);
#pragma unroll
    for (int ns = 0; ns < 4; ++ns) {
        int col = blockN + ns * 16 + l15;
        float bv = bias[col];
#pragma unroll
        for (int j = 0; j < 8; ++j) {
            int row0 = rowBase0 + j;
            if (row0 < RG_N) out[(long)row0 * RG_D + col] = acc0[ns][j] + bv;
            int row1 = rowBase0 + 16 + j;
            if (row1 < RG_N) out[(long)row1 * RG_D + col] = acc1[ns][j] + bv;
        }
    }
}

// ---------------- host-side launch ----------------

extern "C" void kernel_launch(void* const* d_in, const int* in_sizes, int n_in,
                              void* d_out, int out_size, void* d_ws, size_t ws_size,
                              hipStream_t stream) {
    const int*   ei     = (const int*)d_in[0];     // [2, E]
    const int*   et     = (const int*)d_in[1];     // [E]
    const float* emb    = (const float*)d_in[2];   // [N, D]
    const float* comps  = (const float*)d_in[3];   // [L, R, B]
    const float* bases  = (const float*)d_in[4];   // [L, B, D, D]
    const float* roots  = (const float*)d_in[5];   // [L, D, D]
    const float* biases = (const float*)d_in[6];   // [L, D]

    // workspace layout (all 16B aligned)
    char* ws = (char*)d_ws;
    float* cnt  = (float*)ws;                                  // [R*N]
    float* Wbuf = (float*)(ws + (size_t)RG_RN * 4);            // [R*D*D]
    float* agg  = (float*)(ws + (size_t)RG_RN * 4
                              + (size_t)RG_R * RG_D * RG_D * 4); // [R*N, D]
    float* h    = (float*)(ws + (size_t)RG_RN * 4
                              + (size_t)RG_R * RG_D * RG_D * 4
                              + (size_t)RG_RN * RG_D * 4);       // [N, D]
    float* y    = (float*)d_out;

    const long aggN4 = (long)RG_RN * RG_D / 4;
    const dim3 gemmGrid((RG_N + 255) / 256, RG_D / 64);

    // counts -> inverse counts (shared by both layers)
    rgcn_zero_f4<<<1024, 256, 0, stream>>>((float4*)cnt, (long)RG_RN / 4);
    rgcn_count_edges<<<(RG_E + 255) / 256, 256, 0, stream>>>(ei, et, cnt);
    rgcn_invert_cnt<<<(RG_RN + 255) / 256, 256, 0, stream>>>(cnt);

    // ---- layer 0: input = embeddings ----
    rgcn_zero_f4<<<4096, 256, 0, stream>>>((float4*)agg, aggN4);
    rgcn_scatter<<<RG_E, RG_D, 0, stream>>>(ei, et, emb, agg);
    rgcn_compute_w<<<(RG_R * RG_D * RG_D) / 256, 256, 0, stream>>>(
        comps + 0 * RG_R * RG_B, bases + (long)0 * RG_B * RG_D * RG_D, Wbuf);
    rgcn_gemm<<<gemmGrid, 256, 0, stream>>>(
        agg, emb, Wbuf, roots + (long)0 * RG_D * RG_D, biases + 0 * RG_D, cnt, h);

    // ---- layer 1: input = relu(h) ----
    rgcn_relu<<<4096, 256, 0, stream>>>(h, (long)RG_N * RG_D);
    rgcn_zero_f4<<<4096, 256, 0, stream>>>((float4*)agg, aggN4);
    rgcn_scatter<<<RG_E, RG_D, 0, stream>>>(ei, et, h, agg);
    rgcn_compute_w<<<(RG_R * RG_D * RG_D) / 256, 256, 0, stream>>>(
        comps + 1 * RG_R * RG_B, bases + (long)1 * RG_B * RG_D * RG_D, Wbuf);
    rgcn_gemm<<<gemmGrid, 256, 0, stream>>>(
        agg, h, Wbuf, roots + (long)1 * RG_D * RG_D, biases + 1 * RG_D, cnt, y);
}